// SimpleCRF_69080253989583
// MI455X (gfx1250) — compile-verified
//
#include <hip/hip_runtime.h>
#include <hip/hip_bf16.h>
#include <math.h>

// CRF log-likelihood, MI455X (gfx1250, wave32, WMMA).
// Forward recursion as exp-space GEMM: alpha' = log( exp(alpha - m) @ exp(T) ) + m + em
// using V_WMMA_F32_16X16X4_F32 with B (=exp(T) column slab) held in registers.

#define NLAB 128
#define SEQL 512
#define BATCH 128
#define TB   16      // batch rows per workgroup (one WMMA M-tile)
#define PSTR 132     // LDS row stride (floats): 132 % 64 == 4 -> conflict-free A loads

typedef __attribute__((ext_vector_type(2))) float v2f;
typedef __attribute__((ext_vector_type(8))) float v8f;

__launch_bounds__(256)
__global__ void crf_forward_kernel(const float* __restrict__ em,
                                   const unsigned char* __restrict__ mask,
                                   const float* __restrict__ trans,
                                   const float* __restrict__ start_tr,
                                   const float* __restrict__ end_tr,
                                   float* __restrict__ partition) {
  __shared__ float alpha[TB * PSTR];  // current alpha, 16 x 128 (padded)
  __shared__ float pbuf [TB * PSTR];  // exp(alpha - rowmax)
  __shared__ float mrow [TB];         // per-row max

  const int tid  = threadIdx.x;
  const int b0   = blockIdx.x * TB;

  const int wave  = tid >> 5;         // 0..7 -> N-tile
  const int lane  = tid & 31;
  const int lhalf = lane >> 4;        // 0/1 : K-pair select (A) / M-half (C,D)
  const int l16   = lane & 15;        // A: row M ; B/C/D: column N
  const int n0    = wave << 4;

  // ---- Preload B = exp(T) column slab [128 x 16] into registers (loop-invariant) ----
  // B 4x16 layout: lanes 0-15 hold K={0,1} in {v0,v1}; lanes 16-31 hold K={2,3}.
  v2f breg[NLAB / 4];
#pragma unroll
  for (int kk = 0; kk < NLAB / 4; ++kk) {
    const int k = kk * 4 + lhalf * 2;
    breg[kk].x = __expf(trans[(k + 0) * NLAB + n0 + l16]);
    breg[kk].y = __expf(trans[(k + 1) * NLAB + n0 + l16]);
  }

  // ---- alpha0 = emissions[:,0,:] + start_transitions ----
  for (int idx = tid; idx < TB * NLAB; idx += 256) {
    const int r = idx >> 7;
    const int j = idx & (NLAB - 1);
    alpha[r * PSTR + j] = em[(b0 + r) * SEQL * NLAB + j] + start_tr[j];
  }
  __syncthreads();

  const int mr = tid >> 4;            // reduction phase: row 0..15
  const int mc = (tid & 15) << 3;     // 8 columns per thread

  for (int t = 1; t < SEQL; ++t) {
    // ---- phase 1: row max + exp shift (16-lane shuffle reduction, wave32) ----
    float mx = -3.402823466e38f;
#pragma unroll
    for (int i = 0; i < 8; ++i)
      mx = fmaxf(mx, alpha[mr * PSTR + mc + i]);
#pragma unroll
    for (int off = 1; off < 16; off <<= 1)
      mx = fmaxf(mx, __shfl_xor(mx, off, 32));
    if ((tid & 15) == 0) mrow[mr] = mx;
#pragma unroll
    for (int i = 0; i < 8; ++i)
      pbuf[mr * PSTR + mc + i] = __expf(alpha[mr * PSTR + mc + i] - mx);
    __syncthreads();

    // ---- phase 2: S[16 x 16] = P @ E via 32x WMMA_F32_16X16X4_F32 ----
    // 4 independent accumulator chains to hide WMMA latency.
    v8f acc0 = {}; v8f acc1 = {}; v8f acc2 = {}; v8f acc3 = {};
    const float* arow = pbuf + l16 * PSTR + lhalf * 2;  // A 16x4: lane=M, K-pair by half
#pragma unroll
    for (int kk = 0; kk < NLAB / 16; ++kk) {
      const v2f a0 = *(const v2f*)(arow + kk * 16 + 0);
      const v2f a1 = *(const v2f*)(arow + kk * 16 + 4);
      const v2f a2 = *(const v2f*)(arow + kk * 16 + 8);
      const v2f a3 = *(const v2f*)(arow + kk * 16 + 12);
      acc0 = __builtin_amdgcn_wmma_f32_16x16x4_f32(false, a0, false, breg[kk*4+0], (short)0, acc0, false, false);
      acc1 = __builtin_amdgcn_wmma_f32_16x16x4_f32(false, a1, false, breg[kk*4+1], (short)0, acc1, false, false);
      acc2 = __builtin_amdgcn_wmma_f32_16x16x4_f32(false, a2, false, breg[kk*4+2], (short)0, acc2, false, false);
      acc3 = __builtin_amdgcn_wmma_f32_16x16x4_f32(false, a3, false, breg[kk*4+3], (short)0, acc3, false, false);
    }
#pragma unroll
    for (int i = 0; i < 8; ++i) acc0[i] = (acc0[i] + acc1[i]) + (acc2[i] + acc3[i]);

    // ---- phase 3: alpha' = log(S) + m + em (masked); C/D layout: VGPR v -> M=v+8*half ----
#pragma unroll
    for (int v = 0; v < 8; ++v) {
      const int   r   = v + (lhalf << 3);
      const int   gb  = b0 + r;
      const float* ep = &em[(gb * SEQL + t) * NLAB + n0 + l16];
      const float e   = *ep;
      __builtin_prefetch(ep + NLAB, 0, 1);   // next timestep -> global_prefetch_b8
      const float nw  = __logf(acc0[v]) + mrow[r] + e;
      const int   li  = r * PSTR + n0 + l16;
      const float old = alpha[li];
      alpha[li] = mask[gb * SEQL + t] ? nw : old;
    }
    __syncthreads();
  }

  // ---- partition[b] = logsumexp_j(alpha[b,j] + end_transitions[j]) ----
  float vals[8];
  float mx = -3.402823466e38f;
#pragma unroll
  for (int i = 0; i < 8; ++i) {
    vals[i] = alpha[mr * PSTR + mc + i] + end_tr[mc + i];
    mx = fmaxf(mx, vals[i]);
  }
#pragma unroll
  for (int off = 1; off < 16; off <<= 1)
    mx = fmaxf(mx, __shfl_xor(mx, off, 32));
  float s = 0.f;
#pragma unroll
  for (int i = 0; i < 8; ++i) s += __expf(vals[i] - mx);
#pragma unroll
  for (int off = 1; off < 16; off <<= 1)
    s += __shfl_xor(s, off, 32);
  if ((tid & 15) == 0) partition[b0 + mr] = __logf(s) + mx;
}

__launch_bounds__(128)
__global__ void crf_score_kernel(const float* __restrict__ em,
                                 const long long* __restrict__ tags,
                                 const unsigned char* __restrict__ mask,
                                 const float* __restrict__ trans,
                                 const float* __restrict__ start_tr,
                                 const float* __restrict__ end_tr,
                                 const float* __restrict__ partition,
                                 float* __restrict__ out) {
  __shared__ float red[BATCH];
  const int b = threadIdx.x;

  float em_score = 0.f, tr_score = 0.f;
  int prev = (int)tags[b * SEQL];
  int cnt = 0;
  for (int t = 0; t < SEQL; ++t) {
    const int   tg = (int)tags[b * SEQL + t];
    const int   mk = mask[b * SEQL + t] ? 1 : 0;
    const float mf = (float)mk;
    cnt += mk;
    em_score += em[(b * SEQL + t) * NLAB + tg] * mf;
    if (t >= 1) tr_score += trans[prev * NLAB + tg] * mf;
    prev = tg;
  }
  const int last_tag = (int)tags[b * SEQL + (cnt - 1)];
  const float score = em_score + tr_score
                    + start_tr[(int)tags[b * SEQL]] + end_tr[last_tag];
  red[b] = score - partition[b];
  __syncthreads();
  for (int s = BATCH / 2; s > 0; s >>= 1) {
    if (b < s) red[b] += red[b + s];
    __syncthreads();
  }
  if (b == 0) out[0] = red[0] / (float)BATCH;
}

extern "C" void kernel_launch(void* const* d_in, const int* in_sizes, int n_in,
                              void* d_out, int out_size, void* d_ws, size_t ws_size,
                              hipStream_t stream) {
  const float*         em       = (const float*)d_in[0];
  const long long*     tags     = (const long long*)d_in[1];
  const unsigned char* mask     = (const unsigned char*)d_in[2];
  const float*         trans    = (const float*)d_in[3];
  const float*         start_tr = (const float*)d_in[4];
  const float*         end_tr   = (const float*)d_in[5];
  float* partition = (float*)d_ws;   // 128 floats of scratch

  crf_forward_kernel<<<BATCH / TB, 256, 0, stream>>>(em, mask, trans, start_tr, end_tr, partition);
  crf_score_kernel<<<1, BATCH, 0, stream>>>(em, tags, mask, trans, start_tr, end_tr,
                                            partition, (float*)d_out);
}